// MahalanobisLoss_8607114461314
// MI455X (gfx1250) — compile-verified
//
#include <hip/hip_runtime.h>

// ---------------------------------------------------------------------------
// MahalanobisLoss for MI455X (gfx1250, wave32).
//
// Per element: e = pr_x - x;  L = lower-tri(chol);  Q = L L^T
//   md   = e^T Q^-1 e = || L^-1 e ||^2   (forward substitution, no 3x3 inverse)
//   loss = md + log(sum(Q) + 1)
// Output = mean over all B*T elements.
//
// Memory-bound: 201 MB read @ 23.3 TB/s -> ~8.6 us floor. Single coalesced
// pass; deterministic reduction (ws partials + fixed-shape second kernel).
// Wave reduction uses V_WMMA_F32_16X16X4_F32 with A = ones so the matrix
// unit performs the cross-lane sum at full f32 precision.
// ---------------------------------------------------------------------------

typedef float v2f __attribute__((ext_vector_type(2)));
typedef float v8f __attribute__((ext_vector_type(8)));

#define N_TOTAL (65536u * 64u)   // B*T = 4,194,304
#define BLOCKS  4096
#define TPB     256
#define ITERS   (N_TOTAL / (BLOCKS * TPB))   // = 4, exact cover -> no divergence

__global__ __launch_bounds__(TPB) void mahal_partial_kernel(
    const float* __restrict__ pr_x,
    const float* __restrict__ x,
    const float* __restrict__ chol,
    float* __restrict__ partial)
{
    const unsigned tid    = blockIdx.x * TPB + threadIdx.x;
    const unsigned stride = BLOCKS * TPB;

    float acc = 0.0f;
    unsigned idx = tid;
    #pragma unroll
    for (int it = 0; it < (int)ITERS; ++it, idx += stride) {
        if (it + 1 < (int)ITERS) {
            // global_prefetch_b8: pull next grid-stride iteration toward L2
            __builtin_prefetch(pr_x + 3u * (idx + stride), 0, 1);
            __builtin_prefetch(x    + 3u * (idx + stride), 0, 1);
            __builtin_prefetch(chol + 6u * (idx + stride), 0, 1);
        }

        const float* pe = pr_x + 3u * idx;
        const float* xe = x    + 3u * idx;
        const float* ce = chol + 6u * idx;

        const float e0 = pe[0] - xe[0];
        const float e1 = pe[1] - xe[1];
        const float e2 = pe[2] - xe[2];

        const float a = ce[0], b = ce[1], c = ce[2];
        const float d = ce[3], e = ce[4], f = ce[5];

        // Q = L L^T (symmetric); normQ = sum of all 9 entries + 1
        const float Q00 = a * a;
        const float Q01 = a * b;
        const float Q02 = a * d;
        const float Q11 = b * b + c * c;
        const float Q12 = b * d + c * e;
        const float Q22 = d * d + e * e + f * f;
        const float normQ = Q00 + Q11 + Q22 + 2.0f * (Q01 + Q02 + Q12) + 1.0f;

        // e^T Q^-1 e = ||y||^2 where L y = e (forward substitution)
        const float y0 = e0 / a;
        const float y1 = (e1 - b * y0) / c;
        const float y2 = (e2 - d * y0 - e * y1) / f;
        const float md = y0 * y0 + y1 * y1 + y2 * y2;

        acc += md + logf(normQ);
    }

    // ---- wave32 reduction via the matrix unit (full f32 precision) ----
    // A = ones(16x4), B slot0 = per-lane acc, slot1 = 0.
    // D[m][n] = sum_k B[k][n]: the 64 B slots partition into 16 column sums,
    // so one WMMA folds 32 lane accumulators into 16 column partials;
    // 4 xor-shuffles finish the wave total (every lane holds it).
    v2f av; av.x = 1.0f; av.y = 1.0f;
    v2f bv; bv.x = acc;  bv.y = 0.0f;
    v8f cv = {};
    v8f dv = __builtin_amdgcn_wmma_f32_16x16x4_f32(
        /*neg_a=*/false, av, /*neg_b=*/false, bv,
        /*c_mod=*/(short)0, cv, /*reuse_a=*/false, /*reuse_b=*/false);

    float w = dv[0];                 // lane L holds colsum[L % 16]
    w += __shfl_xor(w, 1);
    w += __shfl_xor(w, 2);
    w += __shfl_xor(w, 4);
    w += __shfl_xor(w, 8);           // now: wave total in every lane

    __shared__ float wpart[TPB / 32];
    if ((threadIdx.x & 31u) == 0u) wpart[threadIdx.x >> 5] = w;
    __syncthreads();
    if (threadIdx.x == 0u) {
        float s = 0.0f;
        #pragma unroll
        for (int i = 0; i < TPB / 32; ++i) s += wpart[i];
        partial[blockIdx.x] = s;
    }
}

__global__ __launch_bounds__(256) void mahal_final_kernel(
    const float* __restrict__ partial,
    float* __restrict__ out)
{
    float s = 0.0f;
    for (unsigned i = threadIdx.x; i < (unsigned)BLOCKS; i += 256u)
        s += partial[i];                         // 16 loads/thread, fixed shape

    s += __shfl_xor(s, 1);
    s += __shfl_xor(s, 2);
    s += __shfl_xor(s, 4);
    s += __shfl_xor(s, 8);
    s += __shfl_xor(s, 16);

    __shared__ float wpart[8];
    if ((threadIdx.x & 31u) == 0u) wpart[threadIdx.x >> 5] = s;
    __syncthreads();
    if (threadIdx.x == 0u) {
        float t = 0.0f;
        #pragma unroll
        for (int i = 0; i < 8; ++i) t += wpart[i];
        out[0] = t * (1.0f / (float)N_TOTAL);    // mean over B*T
    }
}

extern "C" void kernel_launch(void* const* d_in, const int* in_sizes, int n_in,
                              void* d_out, int out_size, void* d_ws, size_t ws_size,
                              hipStream_t stream) {
    (void)in_sizes; (void)n_in; (void)out_size; (void)ws_size;
    const float* pr_x = (const float*)d_in[0];
    const float* x    = (const float*)d_in[1];
    const float* chol = (const float*)d_in[2];
    float* partial = (float*)d_ws;               // 4096 floats = 16 KB scratch

    mahal_partial_kernel<<<BLOCKS, TPB, 0, stream>>>(pr_x, x, chol, partial);
    mahal_final_kernel<<<1, 256, 0, stream>>>(partial, (float*)d_out);
}